// DTMLayer_11295763989132
// MI455X (gfx1250) — compile-verified
//
#include <hip/hip_runtime.h>
#include <math.h>

// DTM layer: for each of 25600 grid points, k=41 nearest of 4096 points.
// d^2 = |g|^2 + |p|^2 - 2 g.p ; cross term via V_WMMA_F32_16X16X4_F32 (K=2 padded to 4).
// One wave32 per block owns 32 grid points (two 16-row WMMA tiles).
// Selection: per-lane register-resident sorted top-41; inserts via raw
// v_minmax_num_f32 (inline asm) to avoid IEEE-mode NaN-canonicalization ops.

#pragma float_control(precise, off)

typedef __attribute__((ext_vector_type(2))) float v2f;
typedef __attribute__((ext_vector_type(4))) float v4f;
typedef __attribute__((ext_vector_type(8))) float v8f;

#define NPTS       4096
#define K_SEL      41
#define GRID_PTS   25600
#define ROW_STRIDE 20          // floats; 16B-aligned rows, conflict-free b128 reads
#define BIG        3.4e38f

// dst = max(min(a, b), c)  -- single VOP3 op, no canonicalization inserted.
__device__ __forceinline__ float vminmax(float a, float b, float c) {
    float r;
    asm("v_minmax_num_f32 %0, %1, %2, %3" : "=v"(r) : "v"(a), "v"(b), "v"(c));
    return r;
}
__device__ __forceinline__ float vmin(float a, float b) {
    float r;
    asm("v_min_num_f32 %0, %1, %2" : "=v"(r) : "v"(a), "v"(b));
    return r;
}

// Sorted-descending register list: h[0] = current 41st-smallest (the max kept).
// Insert v and drop h[0]:  h[j] = max(min(h[j], v), h[j+1]).  No-op if v >= h[0].
// All 41 ops read only old values + v, so they are mutually independent.
__device__ __forceinline__ void heap_insert(float (&h)[K_SEL], float v) {
    #pragma unroll
    for (int j = 0; j < K_SEL - 1; ++j)
        h[j] = vminmax(h[j], v, h[j + 1]);
    h[K_SEL - 1] = vmin(h[K_SEL - 1], v);
}

__global__ __launch_bounds__(32)
void dtm_knn_wmma_kernel(const float* __restrict__ x,
                         const float* __restrict__ grid,
                         float* __restrict__ out) {
    __shared__ __align__(16) float tileBuf[32 * ROW_STRIDE]; // 32x16 tile, padded rows

    const int lane    = threadIdx.x;        // 0..31 (wave32)
    const int gBase   = blockIdx.x * 32;    // first grid point of this block
    const int colHalf = lane >> 4;          // selects M-half of C/D tile
    const int col     = lane & 15;          // N index of this lane's tile column

    // This lane's own grid point (row it owns for selection).
    const float gx  = grid[2 * (gBase + lane) + 0];
    const float gy  = grid[2 * (gBase + lane) + 1];
    const float gsq = gx * gx + gy * gy;

    // A matrices (16x4 f32, wave32 layout): lanes 0-15 hold M=0..15 with
    // vgpr0=K0, vgpr1=K1; lanes 16-31 hold K=2,3 which we zero-pad.
    v2f a0 = {0.0f, 0.0f};                  // rows gBase .. gBase+15
    v2f a1 = {0.0f, 0.0f};                  // rows gBase+16 .. gBase+31
    if (lane < 16) {
        const float g0x = grid[2 * (gBase + lane) + 0];
        const float g0y = grid[2 * (gBase + lane) + 1];
        const float g1x = grid[2 * (gBase + 16 + lane) + 0];
        const float g1y = grid[2 * (gBase + 16 + lane) + 1];
        a0.x = -2.0f * g0x;  a0.y = -2.0f * g0y;
        a1.x = -2.0f * g1x;  a1.y = -2.0f * g1y;
    }

    // Register-resident top-41 (sorted descending), init to +inf-ish.
    float heap[K_SEL];
    #pragma unroll
    for (int j = 0; j < K_SEL; ++j) heap[j] = BIG;

    for (int c = 0; c < NPTS / 16; ++c) {
        // All 32 lanes fetch the chunk point for their own tile column, so every
        // lane owns psq of its column and can fold it into the tile at store time.
        const int   p   = c * 16 + col;
        const float px  = x[2 * p + 0];
        const float py  = x[2 * p + 1];
        const float psq = px * px + py * py;

        // B matrix (4x16 f32): lanes 0-15 hold K=0,1 of column N; lanes 16-31 pad K=2,3.
        v2f b;
        b.x = (lane < 16) ? px : 0.0f;
        b.y = (lane < 16) ? py : 0.0f;

        v8f d0 = {};
        v8f d1 = {};
        // D = A x B = -2 * g.p for a 16x16 tile
        d0 = __builtin_amdgcn_wmma_f32_16x16x4_f32(
                 false, a0, false, b, (short)0, d0, false, false);
        d1 = __builtin_amdgcn_wmma_f32_16x16x4_f32(
                 false, a1, false, b, (short)0, d1, false, false);

        // Scatter to LDS as [row][col], folding |p|^2 of this column in.
        // C/D layout: vgpr r -> M = r + 8*(lane>=16), N = lane%16.
        #pragma unroll
        for (int r = 0; r < 8; ++r) {
            tileBuf[(r + 8 * colHalf) * ROW_STRIDE + col]      = d0[r] + psq;
            tileBuf[(16 + r + 8 * colHalf) * ROW_STRIDE + col] = d1[r] + psq;
        }
        __syncthreads();

        // Each lane consumes its own row: issue all 4 b128 loads up front so
        // the LDS latency is paid once per chunk, then 16 candidate checks.
        const v4f rv0 = *(const v4f*)&tileBuf[lane * ROW_STRIDE + 0];
        const v4f rv1 = *(const v4f*)&tileBuf[lane * ROW_STRIDE + 4];
        const v4f rv2 = *(const v4f*)&tileBuf[lane * ROW_STRIDE + 8];
        const v4f rv3 = *(const v4f*)&tileBuf[lane * ROW_STRIDE + 12];

        float cand[16];
        cand[0]  = rv0.x + gsq; cand[1]  = rv0.y + gsq;
        cand[2]  = rv0.z + gsq; cand[3]  = rv0.w + gsq;
        cand[4]  = rv1.x + gsq; cand[5]  = rv1.y + gsq;
        cand[6]  = rv1.z + gsq; cand[7]  = rv1.w + gsq;
        cand[8]  = rv2.x + gsq; cand[9]  = rv2.y + gsq;
        cand[10] = rv2.z + gsq; cand[11] = rv2.w + gsq;
        cand[12] = rv3.x + gsq; cand[13] = rv3.y + gsq;
        cand[14] = rv3.z + gsq; cand[15] = rv3.w + gsq;

        #pragma unroll
        for (int n = 0; n < 16; ++n) {
            if (cand[n] < heap[0]) heap_insert(heap, cand[n]);
        }
        __syncthreads();
    }

    // heap holds the 41 smallest d^2; heap[0] is the 41st smallest.
    float sum = 0.0f;
    #pragma unroll
    for (int j = 0; j < K_SEL; ++j) sum += heap[j];

    const float bound = 0.01f * (float)NPTS;      // 40.96
    const float corr  = bound - (float)K_SEL;     // -0.04
    out[gBase + lane] = sqrtf((sum + heap[0] * corr) / bound);
}

extern "C" void kernel_launch(void* const* d_in, const int* in_sizes, int n_in,
                              void* d_out, int out_size, void* d_ws, size_t ws_size,
                              hipStream_t stream) {
    const float* x    = (const float*)d_in[0];   // (4096, 2) f32
    const float* grid = (const float*)d_in[1];   // (25600, 2) f32
    float* out        = (float*)d_out;           // 25600 f32 (160x160)

    dim3 blocks(GRID_PTS / 32);                  // 800 blocks x 1 wave32
    dtm_knn_wmma_kernel<<<blocks, 32, 0, stream>>>(x, grid, out);
}